// FirstOrderConcat_44659069943968
// MI455X (gfx1250) — compile-verified
//
#include <hip/hip_runtime.h>

// FirstOrderConcat pair-scoring heads for MI455X (gfx1250, wave32).
//
//   K1 proj_kernel : P[z][m][l] = x @ W1[head,side] (+ b1 on side==1), z = head*2+side
//                    fp32 WMMA 16x16x4, 8 GEMMs of 1024x512x512 (4.3 GFLOP)
//   K2 pair_kernel : per (b, i-tile16, j-tile16): h = tanh(L_i + R_j)
//                    scalar heads -> VALU dot with W2 (k split across lane halves,
//                    combined via shfl_xor 16); label head -> fp32 WMMA vs W2_lab.
//                    tanh uses HW v_tanh_f32 when the builtin exists.
//   K3 sym_kernel  : span head triangular symmetrization in place.

typedef __attribute__((ext_vector_type(2))) float v2f;
typedef __attribute__((ext_vector_type(4))) float v4f;
typedef __attribute__((ext_vector_type(8))) float v8f;

#define BB 8
#define SS 128
#define DD 512
#define LL 512
#define RR 50
#define M1 (BB * SS)  // 1024

#if __has_builtin(__builtin_amdgcn_tanhf)
#define TANH(x) __builtin_amdgcn_tanhf(x)
#elif __has_builtin(__builtin_amdgcn_tanh_f32)
#define TANH(x) __builtin_amdgcn_tanh_f32(x)
#else
#define TANH(x) tanhf(x)
#endif

__device__ __forceinline__ v8f wmma_f32(v2f a, v2f b, v8f c) {
  // D = A(16x4 f32) * B(4x16 f32) + C(16x16 f32)
  return __builtin_amdgcn_wmma_f32_16x16x4_f32(false, a, false, b, (short)0, c,
                                               false, false);
}

// ---------------------------------------------------------------------------
// Kernel 1: projections.  4096 waves total: z(8) x mtile(64) x ngroup(8).
// Each wave computes a 16(M) x 64(N) tile of P[z] = X(1024x512) @ W(512x512).
// ---------------------------------------------------------------------------
__global__ __launch_bounds__(256) void proj_kernel(
    const float* __restrict__ x,
    const float* __restrict__ w10, const float* __restrict__ w11,
    const float* __restrict__ w12, const float* __restrict__ w13,
    const float* __restrict__ b10, const float* __restrict__ b11,
    const float* __restrict__ b12, const float* __restrict__ b13,
    float* __restrict__ P) {
  const int lane = threadIdx.x & 31;
  const int jl = lane & 15;   // column within 16 (A row / B,C col)
  const int hi = lane >> 4;   // which k-half of the wave
  const int wid = blockIdx.x * (blockDim.x >> 5) + (threadIdx.x >> 5);
  const int z = wid >> 9;           // 0..7  (head*2 + side)
  const int rem = wid & 511;
  const int mt = rem >> 3;          // 0..63
  const int ng = rem & 7;           // 0..7
  const int head = z >> 1, side = z & 1;

  const float* W1 = head == 0 ? w10 : head == 1 ? w11 : head == 2 ? w12 : w13;
  const float* b1 = head == 0 ? b10 : head == 1 ? b11 : head == 2 ? b12 : b13;
  const float* W = W1 + (size_t)side * DD * LL;  // W[d][l], d = K dim
  const int m0 = mt * 16, n0 = ng * 64;
  const float* xrow = x + (size_t)(m0 + jl) * DD;

  v8f c0 = {}, c1 = {}, c2 = {}, c3 = {};
  for (int k0 = 0; k0 < DD; k0 += 4) {
    const int ka = k0 + 2 * hi;        // this half's first k
    v2f a = *(const v2f*)(xrow + ka);  // A[m=jl][ka], A[m=jl][ka+1]
    const float* wr0 = W + (size_t)ka * LL + n0 + jl;
    const float* wr1 = wr0 + LL;
    v2f bv;
    bv.x = wr0[0];  bv.y = wr1[0];  c0 = wmma_f32(a, bv, c0);
    bv.x = wr0[16]; bv.y = wr1[16]; c1 = wmma_f32(a, bv, c1);
    bv.x = wr0[32]; bv.y = wr1[32]; c2 = wmma_f32(a, bv, c2);
    bv.x = wr0[48]; bv.y = wr1[48]; c3 = wmma_f32(a, bv, c3);
  }

  float* out = P + (size_t)z * M1 * LL;
#pragma unroll
  for (int t = 0; t < 4; t++) {
    v8f c = t == 0 ? c0 : t == 1 ? c1 : t == 2 ? c2 : c3;
    const int col = n0 + t * 16 + jl;
    const float badd = side ? b1[col] : 0.0f;  // fold b1 into right proj
#pragma unroll
    for (int v = 0; v < 8; v++) {
      const int m = m0 + v + 8 * hi;  // C rows: lanes<16 -> M=v, lanes>=16 -> M=v+8
      out[(size_t)m * LL + col] = c[v] + badd;
    }
  }
}

// ---------------------------------------------------------------------------
// Kernel 2: pair stage.  512 workgroups (b x 8 x 8 tiles) x 512 threads.
// Wave w owns row i = i0 + w and all 16 j's of the tile.
// ---------------------------------------------------------------------------
__global__ __launch_bounds__(512) void pair_kernel(
    const float* __restrict__ P,
    const float* __restrict__ w2_0, const float* __restrict__ w2_1,
    const float* __restrict__ w2_2, const float* __restrict__ w2_lab,
    const float* __restrict__ b2_0, const float* __restrict__ b2_1,
    const float* __restrict__ b2_2, const float* __restrict__ b2_lab,
    float* __restrict__ outp) {
  __shared__ float sL[4][16][64];    // 16 KB: left  rows (head, i, k)
  __shared__ float sR[4][16][64];    // 16 KB: right rows (head, j, k) (b1 folded)
  __shared__ v2f sW2lp[32][65];      // 16.25 KB: label W2 chunk as (k,k+1) pairs,
                                     //           padded 50->64 cols + 1 col anti-bank-alias
  __shared__ float sW2s[3][512];     // 6 KB : scalar-head W2 (full K)

  const int tid = threadIdx.x;
  const int lane = tid & 31;
  const int jl = lane & 15;
  const int hi = lane >> 4;
  const int wv = tid >> 5;           // 0..15: i row within tile
  const int b = blockIdx.x >> 6;
  const int rem = blockIdx.x & 63;
  const int i0 = (rem >> 3) * 16;
  const int j0 = (rem & 7) * 16;
  const int i = i0 + wv;

  for (int e = tid; e < 3 * 512; e += 512) {
    const int h = e >> 9, k = e & 511;
    const float* w = h == 0 ? w2_0 : h == 1 ? w2_1 : w2_2;
    sW2s[h][k] = w[k];
  }

  float acc0 = 0.f, acc1 = 0.f, acc2 = 0.f;
  v8f c0 = {}, c1 = {}, c2 = {}, c3 = {};

  for (int kc = 0; kc < 8; kc++) {
    const int kb = kc * 64;
    __syncthreads();
    // Stage left/right rows: 4 heads x 16 rows x 64 k, as float4 (b128).
    for (int e = tid; e < 1024; e += 512) {
      const int h = e >> 8, r = (e >> 4) & 15, k4i = (e & 15) * 4;
      *(v4f*)&sL[h][r][k4i] = *(const v4f*)&P[(size_t)(h * 2 + 0) * M1 * LL +
                                              (size_t)(b * SS + i0 + r) * LL + kb + k4i];
      *(v4f*)&sR[h][r][k4i] = *(const v4f*)&P[(size_t)(h * 2 + 1) * M1 * LL +
                                              (size_t)(b * SS + j0 + r) * LL + kb + k4i];
    }
    // Stage label W2 chunk as k-pairs: sW2lp[kk2][o] = (W2[kb+2*kk2][o], W2[kb+2*kk2+1][o]).
    for (int e = tid; e < 2048; e += 512) {
      const int kk2 = e >> 6, o = e & 63;
      v2f p;
      p.x = (o < RR) ? w2_lab[(size_t)(kb + 2 * kk2) * RR + o] : 0.0f;
      p.y = (o < RR) ? w2_lab[(size_t)(kb + 2 * kk2 + 1) * RR + o] : 0.0f;
      sW2lp[kk2][o] = p;
    }
    __syncthreads();

    for (int it = 0; it < 16; it++) {
      const int kk = it * 4 + 2 * hi;   // lane-half k split (matches WMMA A layout)
      const int row = it * 2 + hi;      // paired-k row in sW2lp
      v2f l0 = *(const v2f*)&sL[0][wv][kk];
      v2f l1 = *(const v2f*)&sL[1][wv][kk];
      v2f l2 = *(const v2f*)&sL[2][wv][kk];
      v2f l3 = *(const v2f*)&sL[3][wv][kk];
      v2f r0 = *(const v2f*)&sR[0][jl][kk];
      v2f r1 = *(const v2f*)&sR[1][jl][kk];
      v2f r2 = *(const v2f*)&sR[2][jl][kk];
      v2f r3 = *(const v2f*)&sR[3][jl][kk];
      const int kg = kb + kk;
      v2f w0 = *(const v2f*)&sW2s[0][kg];
      v2f w1 = *(const v2f*)&sW2s[1][kg];
      v2f w2 = *(const v2f*)&sW2s[2][kg];
      acc0 += TANH(l0.x + r0.x) * w0.x + TANH(l0.y + r0.y) * w0.y;
      acc1 += TANH(l1.x + r1.x) * w1.x + TANH(l1.y + r1.y) * w1.y;
      acc2 += TANH(l2.x + r2.x) * w2.x + TANH(l2.y + r2.y) * w2.y;
      v2f a;
      a.x = TANH(l3.x + r3.x);
      a.y = TANH(l3.y + r3.y);
      c0 = wmma_f32(a, sW2lp[row][jl], c0);
      c1 = wmma_f32(a, sW2lp[row][16 + jl], c1);
      c2 = wmma_f32(a, sW2lp[row][32 + jl], c2);
      c3 = wmma_f32(a, sW2lp[row][48 + jl], c3);
    }
  }

  // Scalar heads: each lane-half holds half the k sum; combine across halves.
  const float s0 = acc0 + __shfl_xor(acc0, 16, 32);
  const float s1 = acc1 + __shfl_xor(acc1, 16, 32);
  const float s2 = acc2 + __shfl_xor(acc2, 16, 32);
  if (lane < 16) {
    const int j = j0 + jl;
    const size_t idx = ((size_t)b * SS + i) * SS + j;
    outp[idx] = s0 + b2_0[0];           // span (raw; symmetrized by K3)
    outp[131072 + idx] = s1 + b2_1[0];  // h2h
    outp[262144 + idx] = s2 + b2_2[0];  // t2t
  }

  // Label head: C[M=j-in-tile][N=o], M = v + 8*hi, N = t*16 + jl.
  float* outl = outp + 393216;
#pragma unroll
  for (int t = 0; t < 4; t++) {
    v8f c = t == 0 ? c0 : t == 1 ? c1 : t == 2 ? c2 : c3;
    const int o = t * 16 + jl;
    if (o < RR) {
      const float badd = b2_lab[o];
#pragma unroll
      for (int v = 0; v < 8; v++) {
        const int j = j0 + v + 8 * hi;
        outl[(((size_t)b * SS + i) * SS + j) * RR + o] = c[v] + badd;
      }
    }
  }
}

// ---------------------------------------------------------------------------
// Kernel 3: span symmetrization.  out[b,i,j] = raw[b,i,j] if i>=j else raw[b,j,i].
// Upper triangle is only written, lower only read -> race-free in place.
// ---------------------------------------------------------------------------
__global__ void sym_kernel(float* __restrict__ outp) {
  const int idx = blockIdx.x * blockDim.x + threadIdx.x;
  if (idx >= BB * SS * SS) return;
  const int j = idx & 127, i = (idx >> 7) & 127, b = idx >> 14;
  if (i < j) outp[idx] = outp[((size_t)b * SS + j) * SS + i];
}

extern "C" void kernel_launch(void* const* d_in, const int* in_sizes, int n_in,
                              void* d_out, int out_size, void* d_ws, size_t ws_size,
                              hipStream_t stream) {
  (void)in_sizes; (void)n_in; (void)out_size; (void)ws_size;
  const float* x       = (const float*)d_in[0];
  const float* span_W1 = (const float*)d_in[1];
  const float* span_b1 = (const float*)d_in[2];
  const float* span_W2 = (const float*)d_in[3];
  const float* span_b2 = (const float*)d_in[4];
  const float* h2h_W1  = (const float*)d_in[5];
  const float* h2h_b1  = (const float*)d_in[6];
  const float* h2h_W2  = (const float*)d_in[7];
  const float* h2h_b2  = (const float*)d_in[8];
  const float* t2t_W1  = (const float*)d_in[9];
  const float* t2t_b1  = (const float*)d_in[10];
  const float* t2t_W2  = (const float*)d_in[11];
  const float* t2t_b2  = (const float*)d_in[12];
  const float* lab_W1  = (const float*)d_in[13];
  const float* lab_b1  = (const float*)d_in[14];
  const float* lab_W2  = (const float*)d_in[15];
  const float* lab_b2  = (const float*)d_in[16];

  float* P = (float*)d_ws;  // 8 * 1024 * 512 floats = 16 MB of scratch
  float* out = (float*)d_out;

  // K1: 4096 waves = 512 blocks x 8 waves.
  proj_kernel<<<512, 256, 0, stream>>>(x, span_W1, h2h_W1, t2t_W1, lab_W1,
                                       span_b1, h2h_b1, t2t_b1, lab_b1, P);
  // K2: 8 b x 8 itiles x 8 jtiles = 512 blocks x 16 waves.
  pair_kernel<<<512, 512, 0, stream>>>(P, span_W2, h2h_W2, t2t_W2, lab_W2,
                                       span_b2, h2h_b2, t2t_b2, lab_b2, out);
  // K3: symmetrize span head.
  sym_kernel<<<(BB * SS * SS + 255) / 256, 256, 0, stream>>>(out);
}